// TripleGrainSeparatePermuter_65901978190083
// MI455X (gfx1250) — compile-verified
//
#include <hip/hip_runtime.h>
#include <stdint.h>

// TripleGrainSeparatePermuter for MI455X (gfx1250, wave32).
// Pure int32 permutation/compaction -> memory-bound. Uses CDNA5 async
// global->LDS loads (ASYNCcnt path) to stage the grain control stream,
// LDS block-scan for stable compaction ranks, and coalesced gather-writes.

#define NCELLS 4096   // N = 64*64 cells per batch
#define FINE_W 256    // fine grid is 256x256
#define TPB    256    // 8 wave32 waves per block

#define CONTENT_PAD 16384
#define CONTENT_EOS 16385
#define CPOS_PAD    4096
#define CPOS_EOS    4097
#define MPOS_PAD    16384
#define MPOS_EOS    16385
#define FPOS_PAD    65536
#define FPOS_EOS    65537

// Flat index into the 256x256 grid for cell n, sub-element q (0..15).
// q = a*8 + b*4 + c*2 + d  maps to row (i1*4 + 2a + c), col (j1*4 + 2b + d).
__device__ __forceinline__ int pos_fine(int n, int q) {
    const int i1 = n >> 6, j1 = n & 63;
    const int a = (q >> 3) & 1, b = (q >> 2) & 1, c = (q >> 1) & 1, d = q & 1;
    return (i1 * 4 + a * 2 + c) * FINE_W + (j1 * 4 + b * 2 + d);
}

// Medium-grid (128x128) position for cell n, sub-element q (0..3).
__device__ __forceinline__ int pos_med(int n, int q) {
    const int i1 = n >> 6, j1 = n & 63;
    const int a = (q >> 1) & 1, b = q & 1;
    return (i1 * 2 + a) * 128 + (j1 * 2 + b);
}

__global__ void tgp_init_ws(int* ws) {
    if (threadIdx.x < 3) ws[threadIdx.x] = 0;
}

// Per-batch class counts -> global maxima in ws[0..2].
__global__ void tgp_count(const int* __restrict__ grain, int* __restrict__ ws) {
    const int b = blockIdx.x;
    const int t = threadIdx.x;
    __shared__ int sc[3];
    if (t < 3) sc[t] = 0;
    __syncthreads();
    int c0 = 0, c1 = 0, c2 = 0;
    const int* g = grain + b * NCELLS;
    for (int n = t; n < NCELLS; n += TPB) {
        const int v = g[n];
        c0 += (v == 0); c1 += (v == 1); c2 += (v == 2);
    }
    atomicAdd(&sc[0], c0); atomicAdd(&sc[1], c1); atomicAdd(&sc[2], c2);
    __syncthreads();
    if (t < 3) atomicMax(&ws[t], sc[t]);
}

// One block per batch: stable-compact the three classes and emit all outputs.
__global__ void tgp_permute(const int* __restrict__ indices,
                            const int* __restrict__ grain,
                            const int* __restrict__ ws,
                            int* __restrict__ out) {
    const int b = blockIdx.x;
    const int t = threadIdx.x;
    const int B = gridDim.x;

    __shared__ __align__(16) int grainLDS[NCELLS];   // 16 KB, staged via async
    __shared__ unsigned short sel[NCELLS];           // rank -> cell, 8 KB
    __shared__ int scan0[TPB + 1], scan1[TPB + 1], scan2[TPB + 1];

    // ---- Stage grain[b] into LDS with CDNA5 async global->LDS loads ----
    {
        const uint32_t ldsBase = (uint32_t)(uintptr_t)grainLDS;  // LDS byte offset
        const int* gsrc = grain + b * NCELLS;
        #pragma unroll
        for (int pass = 0; pass < 4; ++pass) {
            const int idx = pass * 1024 + t * 4;     // int index, 16B aligned
            const unsigned long long ga = (unsigned long long)(uintptr_t)(gsrc + idx);
            const uint32_t la = ldsBase + (uint32_t)(idx * 4);
            asm volatile("global_load_async_to_lds_b128 %0, %1, off"
                         :: "v"(la), "v"(ga) : "memory");
        }
        asm volatile("s_wait_asynccnt 0" ::: "memory");
    }
    __syncthreads();

    // ---- Per-thread counts over a contiguous 16-cell chunk ----
    const int base = t * 16;
    int l0 = 0, l1 = 0, l2 = 0;
    #pragma unroll
    for (int k = 0; k < 16; ++k) {
        const int v = grainLDS[base + k];
        l0 += (v == 0); l1 += (v == 1); l2 += (v == 2);
    }
    scan0[t + 1] = l0; scan1[t + 1] = l1; scan2[t + 1] = l2;
    if (t == 0) { scan0[0] = 0; scan1[0] = 0; scan2[0] = 0; }
    __syncthreads();

    // ---- Hillis-Steele inclusive scan over the 256 per-thread counts ----
    for (int off = 1; off < TPB; off <<= 1) {
        int v0 = 0, v1 = 0, v2 = 0;
        if (t >= off) { v0 = scan0[t + 1 - off]; v1 = scan1[t + 1 - off]; v2 = scan2[t + 1 - off]; }
        __syncthreads();
        if (t >= off) { scan0[t + 1] += v0; scan1[t + 1] += v1; scan2[t + 1] += v2; }
        __syncthreads();
    }
    const int cnt0 = scan0[TPB], cnt1 = scan1[TPB], cnt2 = scan2[TPB];

    // ---- Build stable rank->cell lists (class0 | class1 | class2) ----
    {
        int r0 = scan0[t];
        int r1 = cnt0 + scan1[t];
        int r2 = cnt0 + cnt1 + scan2[t];
        #pragma unroll
        for (int k = 0; k < 16; ++k) {
            const int n = base + k;
            const int v = grainLDS[n];
            if (v == 0)      sel[r0++] = (unsigned short)n;
            else if (v == 1) sel[r1++] = (unsigned short)n;
            else             sel[r2++] = (unsigned short)n;
        }
    }
    __syncthreads();

    // ---- Output geometry from global maxima ----
    const int c0m = ws[0], c1m = ws[1], c2m = ws[2];
    const int Lcc = c0m + 1;          // coarse content / position / segment width
    const int Lmc = 16 * c1m + 1;     // medium content / segment width
    const int Lmp = 4  * c1m + 1;     // medium position width
    const int Lfc = 16 * c2m + 1;     // fine content / position / segment width

    const int base_cc = 0;
    const int base_mc = base_cc + B * Lcc;
    const int base_fc = base_mc + B * Lmc;
    const int base_cp = base_fc + B * Lfc;
    const int base_mp = base_cp + B * Lcc;
    const int base_fp = base_mp + B * Lmp;
    const int base_cs = base_fp + B * Lfc;
    const int base_ms = base_cs + B * Lcc;
    const int base_fs = base_ms + B * Lmc;

    const int* __restrict__ flat = indices + b * (FINE_W * FINE_W);

    // ---- Coarse: content, position, segment (k = 1) ----
    for (int j = t; j < Lcc; j += TPB) {
        int cc, cp;
        if (j < cnt0)       { const int n = sel[j]; cc = flat[pos_fine(n, 0)]; cp = n; }
        else if (j == cnt0) { cc = CONTENT_EOS; cp = CPOS_EOS; }
        else                { cc = CONTENT_PAD; cp = CPOS_PAD; }
        out[base_cc + b * Lcc + j] = cc;
        out[base_cp + b * Lcc + j] = cp;
        out[base_cs + b * Lcc + j] = 0;
    }

    // ---- Medium content + segment (k = 16) ----
    {
        const int lim = cnt1 * 16;
        for (int j = t; j < Lmc; j += TPB) {
            int v;
            if (j < lim) {
                const int n = sel[cnt0 + (j >> 4)];
                v = flat[pos_fine(n, j & 15)];
            } else {
                v = (j == lim) ? CONTENT_EOS : CONTENT_PAD;
            }
            out[base_mc + b * Lmc + j] = v;
            out[base_ms + b * Lmc + j] = 1;
        }
    }

    // ---- Medium position (k = 4) ----
    {
        const int lim = cnt1 * 4;
        for (int j = t; j < Lmp; j += TPB) {
            int v;
            if (j < lim) {
                const int n = sel[cnt0 + (j >> 2)];
                v = pos_med(n, j & 3);
            } else {
                v = (j == lim) ? MPOS_EOS : MPOS_PAD;
            }
            out[base_mp + b * Lmp + j] = v;
        }
    }

    // ---- Fine content, position, segment (k = 16) ----
    {
        const int lim = cnt2 * 16;
        const int sbase = cnt0 + cnt1;
        for (int j = t; j < Lfc; j += TPB) {
            int vc, vp;
            if (j < lim) {
                const int n = sel[sbase + (j >> 4)];
                const int p = pos_fine(n, j & 15);
                vc = flat[p];
                vp = p;
            } else {
                vc = (j == lim) ? CONTENT_EOS : CONTENT_PAD;
                vp = (j == lim) ? FPOS_EOS    : FPOS_PAD;
            }
            out[base_fc + b * Lfc + j] = vc;
            out[base_fp + b * Lfc + j] = vp;
            out[base_fs + b * Lfc + j] = 2;
        }
    }
}

extern "C" void kernel_launch(void* const* d_in, const int* in_sizes, int n_in,
                              void* d_out, int out_size, void* d_ws, size_t ws_size,
                              hipStream_t stream) {
    const int* indices = (const int*)d_in[0];   // (B, 256, 256) int32
    const int* grain   = (const int*)d_in[1];   // (B, 64, 64)   int32
    int* out = (int*)d_out;
    int* ws  = (int*)d_ws;

    const int B = in_sizes[1] / NCELLS;

    tgp_init_ws<<<1, 32, 0, stream>>>(ws);
    tgp_count  <<<B, TPB, 0, stream>>>(grain, ws);
    tgp_permute<<<B, TPB, 0, stream>>>(indices, grain, ws, out);
}